// WelfordEstimator_23235773071887
// MI455X (gfx1250) — compile-verified
//
#include <hip/hip_runtime.h>
#include <stdint.h>

typedef float v4f __attribute__((ext_vector_type(4)));
typedef int   v4i __attribute__((ext_vector_type(4)));

#define B_DIM 32
#define S_DIM 2048
#define H_DIM 1024
#define SH    (S_DIM * H_DIM)        /* 2,097,152 */
#define XTOT  ((size_t)B_DIM * SH)   /* 67,108,864 */
#define BLOCK 256

// ---- CDNA5 async global->LDS path (gfx1250) --------------------------------
#if defined(__gfx1250__) && __has_builtin(__builtin_amdgcn_global_load_async_to_lds_b128)
#define USE_ASYNC 1
#else
#define USE_ASYNC 0
#endif

#if defined(__gfx1250__) && __has_builtin(__builtin_amdgcn_s_wait_asynccnt)
#define WAIT_ASYNC(n) __builtin_amdgcn_s_wait_asynccnt(n)
#else
#define WAIT_ASYNC(n) asm volatile("s_wait_asynccnt " #n ::: "memory")
#endif

#define WAIT_DS0() asm volatile("s_wait_dscnt 0x0" ::: "memory")

#if USE_ASYNC
// Builtin signature (from diagnostics): non-const typed vector pointers,
// (v4i addrspace(1)*, v4i addrspace(3)*, imm offset, imm cpol).
typedef __attribute__((address_space(1))) v4i global_v4i;
typedef __attribute__((address_space(3))) v4i lds_v4i;
#define ASYNC_LOAD_B128(gsrc, ldst)                                        \
  __builtin_amdgcn_global_load_async_to_lds_b128(                          \
      (global_v4i*)(gsrc),                                                 \
      (lds_v4i*)(ldst),                                                    \
      0, 0)
#endif

__global__ __launch_bounds__(BLOCK) void welford_fused_kernel(
    const float* __restrict__ x,
    const float* __restrict__ mean_in,
    const float* __restrict__ m2_in,
    const int*   __restrict__ nz_in,
    const int*   __restrict__ nsamp,
    float*       __restrict__ out)
{
    const int    tid  = threadIdx.x;
    const size_t gid  = (size_t)blockIdx.x * BLOCK + tid;  // float4 index into S*H
    const size_t base = gid * 4;

    float* __restrict__ out_x    = out;
    float* __restrict__ out_mean = out + XTOT;
    float* __restrict__ out_m2   = out + XTOT + (size_t)SH;
    int*   __restrict__ out_nz   = (int*)(out + XTOT + 2 * (size_t)SH);
    int*   __restrict__ out_n    = (int*)(out + XTOT + 3 * (size_t)SH);

    const int n0 = nsamp[0];

    // Running state: 4 independent Welford columns per thread.
    v4f mean = *(const v4f*)(mean_in + base);
    v4f m2v  = *(const v4f*)(m2_in  + base);
    v4i nzv  = *(const v4i*)(nz_in  + base);

#if USE_ASYNC
    __shared__ v4f lbuf[2][BLOCK];   // 8 KB double buffer per block

    // Prologue: prefetch batch slices 0 and 1 into LDS (128b per lane).
    ASYNC_LOAD_B128(x + base,              &lbuf[0][tid]);
    ASYNC_LOAD_B128(x + (size_t)SH + base, &lbuf[1][tid]);
#endif

#pragma unroll
    for (int b = 0; b < B_DIM; ++b) {
        v4f v;
#if USE_ASYNC
        // Oldest outstanding async load is slice b; drain to <=1 (or 0 at the tail).
        if (b == B_DIM - 1) { WAIT_ASYNC(0); } else { WAIT_ASYNC(1); }
        v = lbuf[b & 1][tid];
        // Make sure the ds_load has sampled LDS before the async engine
        // overwrites this buffer with slice b+2.
        WAIT_DS0();
        if (b + 2 < B_DIM)
            ASYNC_LOAD_B128(x + (size_t)(b + 2) * SH + base, &lbuf[b & 1][tid]);
#else
        // Fallback: direct streaming (non-temporal) 128-bit load.
        v = __builtin_nontemporal_load((const v4f*)(x + (size_t)b * SH + base));
#endif

        // Welford step (division replaced by reciprocal multiply).
        const float inv = 1.0f / (float)(n0 + b + 1);
#pragma unroll
        for (int e = 0; e < 4; ++e) {
            const float xi = v[e];
            nzv[e] += (xi != 0.0f) ? 1 : 0;
            const float d = xi - mean[e];          // xi - old_mean
            mean[e] += d * inv;
            m2v[e]  += (xi - mean[e]) * d;         // (xi-new_mean)*(xi-old_mean)
        }

        // Fused passthrough copy of x (write-once -> non-temporal).
        __builtin_nontemporal_store(v, (v4f*)(out_x + (size_t)b * SH + base));
    }

    // Final state (write-once -> non-temporal).
    __builtin_nontemporal_store(mean, (v4f*)(out_mean + base));
    __builtin_nontemporal_store(m2v,  (v4f*)(out_m2   + base));
    __builtin_nontemporal_store(nzv,  (v4i*)(out_nz   + base));

    if (gid == 0)
        out_n[0] = n0 + B_DIM;
}

extern "C" void kernel_launch(void* const* d_in, const int* in_sizes, int n_in,
                              void* d_out, int out_size, void* d_ws, size_t ws_size,
                              hipStream_t stream) {
    (void)in_sizes; (void)n_in; (void)out_size; (void)d_ws; (void)ws_size;

    const float* x    = (const float*)d_in[0];
    const float* mean = (const float*)d_in[1];
    const float* m2   = (const float*)d_in[2];
    const int*   nz   = (const int*)d_in[3];
    const int*   n    = (const int*)d_in[4];
    float*       out  = (float*)d_out;

    const int blocks = SH / 4 / BLOCK;  // 2048
    welford_fused_kernel<<<dim3(blocks), dim3(BLOCK), 0, stream>>>(
        x, mean, m2, nz, n, out);
}